// SyMFM6D_68221260530192
// MI455X (gfx1250) — compile-verified
//
#include <hip/hip_runtime.h>

// ---------------------------------------------------------------------------
// SyMFM6D point<->pixel fusion for MI455X (gfx1250, wave32, WMMA).
// fp32 WMMA (V_WMMA_F32_16X16X4_F32) for the channel matmuls; gathers and
// BN+ReLU fused into prologue/epilogue. Bandwidth-bound (~200MB @ 23.3TB/s).
// ---------------------------------------------------------------------------

typedef __attribute__((ext_vector_type(2))) float v2f;
typedef __attribute__((ext_vector_type(8))) float v8f;

__device__ __forceinline__ v8f wmma_f32(v2f a, v2f b, v8f c) {
  // 8-arg form: (neg_a, A, neg_b, B, c_mod, C, reuse_a, reuse_b)
  return __builtin_amdgcn_wmma_f32_16x16x4_f32(false, a, false, b,
                                               (short)0, c, false, false);
}

// ---------------------------------------------------------------------------
// Gather K=16 neighbor features and max-reduce.
// feat [B,64,Nin], idx [B,M,16] -> out [B,64,M]
// ---------------------------------------------------------------------------
__global__ void gather_maxpool_kernel(const float* __restrict__ feat,
                                      const int* __restrict__ idx,
                                      float* __restrict__ out,
                                      int Nin, int M) {
  const int b = blockIdx.y;
  const float* fb = feat + (size_t)b * 64 * Nin;
  const int*   ib = idx  + (size_t)b * M * 16;
  float*       ob = out  + (size_t)b * 64 * M;
  const int total = 64 * M;
  for (int e = blockIdx.x * blockDim.x + threadIdx.x; e < total;
       e += gridDim.x * blockDim.x) {
    const int c = e / M;
    const int m = e - c * M;
    const float* fr = fb + (size_t)c * Nin;
    const int*   ip = ib + (size_t)m * 16;
    float v = fr[ip[0]];
#pragma unroll
    for (int k = 1; k < 16; ++k) v = fmaxf(v, fr[ip[k]]);
    ob[e] = v;  // coalesced: m is fastest
  }
}

// ---------------------------------------------------------------------------
// Y[64, cols] = relu( scale .* (W[64,CIN] @ X[CIN, cols]) + bias )
//   CIN=64 : X = X0 only.
//   CIN=128: channels 0..63 from X0[col], channels 64..127 from
//            X1[idx ? idx[col] : col]   (fused concat + nearest-interp).
// One wave -> 64 out-channels x 32 columns (4 M-tiles x 2 N-tiles),
// K accumulated with V_WMMA_F32_16X16X4_F32. 8 waves (256 thr) per block.
// ---------------------------------------------------------------------------
template <int CIN>
__global__ void conv_wmma_kernel(const float* __restrict__ X0,
                                 const float* __restrict__ X1,
                                 const int* __restrict__ idx,
                                 const float* __restrict__ W,
                                 const float* __restrict__ scale,
                                 const float* __restrict__ bias,
                                 float* __restrict__ out,
                                 int N, int ld0, int ld1, int ldo, int ocol0) {
  const int b    = blockIdx.y;
  const int lane = threadIdx.x & 31;
  const int wave = threadIdx.x >> 5;
  const int n0   = (blockIdx.x * 8 + wave) * 32;
  if (n0 >= N) return;  // wave-uniform: EXEC stays all-ones for WMMA

  const float* x0 = X0 + (size_t)b * 64 * ld0;
  const float* x1 = (CIN == 128) ? (X1 + (size_t)b * 64 * ld1) : nullptr;
  float*       ob = out + (size_t)b * 64 * ldo;

  const int half = lane >> 4;    // lo/hi lane half
  const int lcol = lane & 15;
  const int c0 = n0 + lcol;      // column, N-tile 0
  const int c1 = c0 + 16;        // column, N-tile 1
  int g0 = c0, g1 = c1;          // gather columns for X1
  if (CIN == 128 && idx != nullptr) {
    const int* ib = idx + (size_t)b * N;
    g0 = ib[c0];
    g1 = ib[c1];
  }

  v8f acc[4][2] = {};

#pragma unroll
  for (int kk = 0; kk < CIN / 4; ++kk) {
    const int k0 = kk * 4;
    const int kb = k0 + 2 * half;  // K rows held by this lane half
    // ---- B fragments (4x16 activation tiles), per ISA half-split layout ----
    v2f bf0, bf1;
    if (CIN == 64 || k0 < 64) {
      bf0[0] = x0[(size_t)kb * ld0 + c0];
      bf0[1] = x0[(size_t)(kb + 1) * ld0 + c0];
      bf1[0] = x0[(size_t)kb * ld0 + c1];
      bf1[1] = x0[(size_t)(kb + 1) * ld0 + c1];
    } else {
      const int kc = kb - 64;
      bf0[0] = x1[(size_t)kc * ld1 + g0];
      bf0[1] = x1[(size_t)(kc + 1) * ld1 + g0];
      bf1[0] = x1[(size_t)kc * ld1 + g1];
      bf1[1] = x1[(size_t)(kc + 1) * ld1 + g1];
    }
    // ---- A fragments (16x4 weight tiles) + 8 WMMAs ----
#pragma unroll
    for (int mt = 0; mt < 4; ++mt) {
      const int row = mt * 16 + lcol;  // output channel row
      v2f a = *(const v2f*)(W + (size_t)row * CIN + kb);  // 8B-aligned
      acc[mt][0] = wmma_f32(a, bf0, acc[mt][0]);
      acc[mt][1] = wmma_f32(a, bf1, acc[mt][1]);
    }
  }

  // ---- epilogue: folded BN + ReLU, direct store (C/D layout: VGPR r holds
  //      rows {mt*16+r, mt*16+r+8} split across lane halves) ----
#pragma unroll
  for (int mt = 0; mt < 4; ++mt) {
#pragma unroll
    for (int r = 0; r < 8; ++r) {
      const int ch = mt * 16 + r + 8 * half;
      const float sv = scale[ch];
      const float bv = bias[ch];
      const float y0 = fmaxf(acc[mt][0][r] * sv + bv, 0.0f);
      const float y1 = fmaxf(acc[mt][1][r] * sv + bv, 0.0f);
      float* orow = ob + (size_t)ch * ldo + ocol0;
      orow[c0] = y0;
      orow[c1] = y1;
    }
  }
}

// ---------------------------------------------------------------------------
// Launch
// ---------------------------------------------------------------------------
extern "C" void kernel_launch(void* const* d_in, const int* in_sizes, int n_in,
                              void* d_out, int out_size, void* d_ws,
                              size_t ws_size, hipStream_t stream) {
  const float* rgb        = (const float*)d_in[0];   // [8,64,120,160]
  const float* pfeat      = (const float*)d_in[1];   // [8,64,12800]
  const float* w_p2r_pre  = (const float*)d_in[2];
  const float* s_p2r_pre  = (const float*)d_in[3];
  const float* b_p2r_pre  = (const float*)d_in[4];
  const float* w_p2r_fuse = (const float*)d_in[5];   // [64,128]
  const float* s_p2r_fuse = (const float*)d_in[6];
  const float* b_p2r_fuse = (const float*)d_in[7];
  const float* w_r2p_pre  = (const float*)d_in[8];
  const float* s_r2p_pre  = (const float*)d_in[9];
  const float* b_r2p_pre  = (const float*)d_in[10];
  const float* w_r2p_fuse = (const float*)d_in[11];  // [64,128]
  const float* s_r2p_fuse = (const float*)d_in[12];
  const float* b_r2p_fuse = (const float*)d_in[13];
  const int*   pool_idx   = (const int*)d_in[14];    // [8,3200,16] -> 12800
  const int*   p2r_idx    = (const int*)d_in[15];    // [8,19200,1] -> 3200
  const int*   r2p_idx    = (const int*)d_in[16];    // [8,3200,16] -> 19200
  float* out = (float*)d_out;                        // [8,64,22400]

  const int Bn = 8, HW = 19200, NS = 3200, NP = 12800, OUTN = 22400;
  float* ws = (float*)d_ws;
  const size_t SZ = (size_t)Bn * 64 * NS;  // 1,638,400 floats
  float* p_emb0 = ws;           // [8,64,3200]
  float* p2r    = ws + SZ;      // [8,64,3200]
  float* r2p_g  = ws + 2 * SZ;  // [8,64,3200]
  float* r2p    = ws + 3 * SZ;  // [8,64,3200]

  const dim3 blk(256);

  // 1) point branch downsample: gather kNN + maxpool
  gather_maxpool_kernel<<<dim3(800, Bn), blk, 0, stream>>>(
      pfeat, pool_idx, p_emb0, NP, NS);

  // 2) p2r pre-conv (64x64) + BN + ReLU
  conv_wmma_kernel<64><<<dim3(13, Bn), blk, 0, stream>>>(
      p_emb0, nullptr, nullptr, w_p2r_pre, s_p2r_pre, b_p2r_pre,
      p2r, NS, NS, 0, NS, 0);

  // 3) rgb fuse: concat(rgb, nearest(p2r)) -> 64x128 conv -> out[:, :, 0:HW]
  conv_wmma_kernel<128><<<dim3(75, Bn), blk, 0, stream>>>(
      rgb, p2r, p2r_idx, w_p2r_fuse, s_p2r_fuse, b_p2r_fuse,
      out, HW, HW, NS, OUTN, 0);

  // 4) r2p gather+maxpool from rgb
  gather_maxpool_kernel<<<dim3(800, Bn), blk, 0, stream>>>(
      rgb, r2p_idx, r2p_g, HW, NS);

  // 5) r2p pre-conv (64x64) + BN + ReLU
  conv_wmma_kernel<64><<<dim3(13, Bn), blk, 0, stream>>>(
      r2p_g, nullptr, nullptr, w_r2p_pre, s_r2p_pre, b_r2p_pre,
      r2p, NS, NS, 0, NS, 0);

  // 6) point fuse: concat(p_emb0, r2p) -> 64x128 conv -> out[:, :, HW:]
  conv_wmma_kernel<128><<<dim3(13, Bn), blk, 0, stream>>>(
      p_emb0, r2p, nullptr, w_r2p_fuse, s_r2p_fuse, b_r2p_fuse,
      out, NS, NS, NS, OUTN, HW);
}